// MultiScaleDeformableKeypointSampler_68925635166332
// MI455X (gfx1250) — compile-verified
//
#include <hip/hip_runtime.h>
#include <math.h>

typedef __attribute__((ext_vector_type(2))) float v2f;
typedef __attribute__((ext_vector_type(8))) float v8f;

#define BT   128
#define CC   256
#define HH   64
#define WWD  64
#define JJ   17
#define HP   32
#define ROWS (BT * JJ)   // 2176 = 136 * 16

// ---------------------------------------------------------------------------
// Kernel 1: q[row, c] = bilinear(feat[b, c], ref[row])   row = b*J + j
// ---------------------------------------------------------------------------
__global__ void k_sample_q(const float* __restrict__ feat,
                           const float* __restrict__ ref,
                           float* __restrict__ q) {
  int row = blockIdx.x;          // 0..2175
  int c   = threadIdx.x;         // 0..255
  int b   = row / JJ;

  float gx = ref[row * 2 + 0];
  float gy = ref[row * 2 + 1];
  float ix = ((gx + 1.f) * (float)WWD - 1.f) * 0.5f;
  float iy = ((gy + 1.f) * (float)HH  - 1.f) * 0.5f;
  float x0f = floorf(ix), y0f = floorf(iy);
  int   x0 = (int)x0f,    y0 = (int)y0f;
  int   x1 = x0 + 1,      y1 = y0 + 1;
  float wx1 = ix - x0f, wy1 = iy - y0f;
  float wx0 = 1.f - wx1, wy0 = 1.f - wy1;
  float vx0 = (x0 >= 0 && x0 < WWD) ? 1.f : 0.f;
  float vx1 = (x1 >= 0 && x1 < WWD) ? 1.f : 0.f;
  float vy0 = (y0 >= 0 && y0 < HH)  ? 1.f : 0.f;
  float vy1 = (y1 >= 0 && y1 < HH)  ? 1.f : 0.f;
  int x0c = min(max(x0, 0), WWD - 1), x1c = min(max(x1, 0), WWD - 1);
  int y0c = min(max(y0, 0), HH  - 1), y1c = min(max(y1, 0), HH  - 1);

  const float* fb = feat + ((size_t)(b * CC + c)) * (HH * WWD);
  float v = wx0 * wy0 * vx0 * vy0 * fb[y0c * WWD + x0c]
          + wx1 * wy0 * vx1 * vy0 * fb[y0c * WWD + x1c]
          + wx0 * wy1 * vx0 * vy1 * fb[y1c * WWD + x0c]
          + wx1 * wy1 * vx1 * vy1 * fb[y1c * WWD + x1c];
  q[(size_t)row * CC + c] = v;
}

// ---------------------------------------------------------------------------
// Kernel 2: S = q @ [W_off | W_w] + [b_off | b_w]   (fp32 WMMA 16x16x4)
//   S layout: [ROWS, 96]  cols 0..63 = offsets, cols 64..95 = weight logits
//   grid: (136, 6) tiles, 1 wave (32 thr) per 16x16 output tile
// ---------------------------------------------------------------------------
__global__ void k_qproj_wmma(const float* __restrict__ q,
                             const float* __restrict__ Woff,
                             const float* __restrict__ boff,
                             const float* __restrict__ Ww,
                             const float* __restrict__ bw,
                             float* __restrict__ S) {
  int tm   = blockIdx.x;           // 0..135 (M tiles)
  int tn   = blockIdx.y;           // 0..5   (N tiles over concat 96)
  int lane = threadIdx.x;          // 0..31
  int nloc = lane & 15;
  int half = lane >> 4;            // 0: K lanes 0/1, 1: K lanes 2/3

  const float* Wsel; const float* bsel; int ldw; int colLocal;
  if (tn < 4) { Wsel = Woff; bsel = boff; ldw = 64; colLocal = tn * 16; }
  else        { Wsel = Ww;   bsel = bw;   ldw = 32; colLocal = (tn - 4) * 16; }

  int m    = tm * 16 + nloc;       // A-fragment row for this lane
  int ncol = colLocal + nloc;      // B-fragment column for this lane

  const float* arow = q + (size_t)m * CC;
  v8f acc = {};
  #pragma unroll 8
  for (int kb = 0; kb < CC; kb += 4) {
    v2f a, bf;
    a.x  = arow[kb + half * 2 + 0];
    a.y  = arow[kb + half * 2 + 1];
    bf.x = Wsel[(size_t)(kb + half * 2 + 0) * ldw + ncol];
    bf.y = Wsel[(size_t)(kb + half * 2 + 1) * ldw + ncol];
    acc = __builtin_amdgcn_wmma_f32_16x16x4_f32(
        /*neg_a=*/false, a, /*neg_b=*/false, bf,
        /*c_mod=*/(short)0, acc, /*reuse_a=*/false, /*reuse_b=*/false);
  }

  float bias = bsel[ncol];
  #pragma unroll
  for (int r = 0; r < 8; ++r) {
    int mrow = tm * 16 + r + half * 8;       // C/D layout: row = vgpr + 8*half
    S[(size_t)mrow * 96 + tn * 16 + nloc] = acc[r] + bias;
  }
}

// ---------------------------------------------------------------------------
// Kernel 3: softmax over 32 weight logits (wave-shuffle reduce) + build grid
// ---------------------------------------------------------------------------
__global__ void k_softmax_grid(const float* __restrict__ S,
                               const float* __restrict__ ref,
                               float* __restrict__ grid,
                               float* __restrict__ wts) {
  int lane = threadIdx.x & 31;
  int row  = blockIdx.x * (blockDim.x >> 5) + (threadIdx.x >> 5);
  if (row >= ROWS) return;

  float x  = S[(size_t)row * 96 + 64 + lane];
  float mx = x;
  #pragma unroll
  for (int off = 16; off > 0; off >>= 1) mx = fmaxf(mx, __shfl_xor(mx, off, 32));
  float e = __expf(x - mx);
  float s = e;
  #pragma unroll
  for (int off = 16; off > 0; off >>= 1) s += __shfl_xor(s, off, 32);
  wts[(size_t)row * HP + lane] = e / s;

  float rx = ref[row * 2 + 0], ry = ref[row * 2 + 1];
  float ox = S[(size_t)row * 96 + 2 * lane + 0];
  float oy = S[(size_t)row * 96 + 2 * lane + 1];
  grid[(size_t)row * 64 + 2 * lane + 0] = fminf(fmaxf(rx + ox, -1.f), 1.f);
  grid[(size_t)row * 64 + 2 * lane + 1] = fminf(fmaxf(ry + oy, -1.f), 1.f);
}

// ---------------------------------------------------------------------------
// Kernel 4: agg[row, c] = sum_p wts[row,p] * bilinear(feat[b,c], grid[row,p])
//   Block per row; 32 lanes precompute corner idx + attention-scaled corner
//   weights into LDS; 256 channel threads do 32x4 gathers each.
// ---------------------------------------------------------------------------
__global__ void k_deform_agg(const float* __restrict__ feat,
                             const float* __restrict__ grid,
                             const float* __restrict__ wts,
                             float* __restrict__ agg) {
  __shared__ int   sx0[HP], sx1[HP], sy0[HP], sy1[HP];
  __shared__ float sw00[HP], sw01[HP], sw10[HP], sw11[HP];

  int row = blockIdx.x;
  int c   = threadIdx.x;
  int b   = row / JJ;

  if (c < HP) {
    int p = c;
    float gx = grid[(size_t)row * 64 + 2 * p + 0];
    float gy = grid[(size_t)row * 64 + 2 * p + 1];
    float w  = wts[(size_t)row * HP + p];
    float ix = ((gx + 1.f) * (float)WWD - 1.f) * 0.5f;
    float iy = ((gy + 1.f) * (float)HH  - 1.f) * 0.5f;
    float x0f = floorf(ix), y0f = floorf(iy);
    int   x0 = (int)x0f,    y0 = (int)y0f;
    int   x1 = x0 + 1,      y1 = y0 + 1;
    float wx1 = ix - x0f, wy1 = iy - y0f;
    float wx0 = 1.f - wx1, wy0 = 1.f - wy1;
    float vx0 = (x0 >= 0 && x0 < WWD) ? 1.f : 0.f;
    float vx1 = (x1 >= 0 && x1 < WWD) ? 1.f : 0.f;
    float vy0 = (y0 >= 0 && y0 < HH)  ? 1.f : 0.f;
    float vy1 = (y1 >= 0 && y1 < HH)  ? 1.f : 0.f;
    sx0[p] = min(max(x0, 0), WWD - 1);
    sx1[p] = min(max(x1, 0), WWD - 1);
    sy0[p] = min(max(y0, 0), HH - 1);
    sy1[p] = min(max(y1, 0), HH - 1);
    sw00[p] = w * wx0 * wy0 * vx0 * vy0;
    sw01[p] = w * wx1 * wy0 * vx1 * vy0;
    sw10[p] = w * wx0 * wy1 * vx0 * vy1;
    sw11[p] = w * wx1 * wy1 * vx1 * vy1;
  }
  __syncthreads();

  const float* fb = feat + ((size_t)(b * CC + c)) * (HH * WWD);
  float acc = 0.f;
  #pragma unroll 4
  for (int p = 0; p < HP; ++p) {
    int y0o = sy0[p] * WWD, y1o = sy1[p] * WWD;
    acc += sw00[p] * fb[y0o + sx0[p]] + sw01[p] * fb[y0o + sx1[p]]
         + sw10[p] * fb[y1o + sx0[p]] + sw11[p] * fb[y1o + sx1[p]];
  }
  agg[(size_t)row * CC + c] = acc;
}

// ---------------------------------------------------------------------------
// Kernel 5: out = agg @ W_out + b_out   (2176 x 256 x 256, fp32 WMMA)
// ---------------------------------------------------------------------------
__global__ void k_outproj_wmma(const float* __restrict__ agg,
                               const float* __restrict__ Wout,
                               const float* __restrict__ bout,
                               float* __restrict__ out) {
  int tm   = blockIdx.x;          // 0..135
  int tn   = blockIdx.y;          // 0..15
  int lane = threadIdx.x;
  int nloc = lane & 15;
  int half = lane >> 4;
  int m    = tm * 16 + nloc;
  int ncol = tn * 16 + nloc;

  const float* arow = agg + (size_t)m * CC;
  v8f acc = {};
  #pragma unroll 8
  for (int kb = 0; kb < CC; kb += 4) {
    v2f a, bf;
    a.x  = arow[kb + half * 2 + 0];
    a.y  = arow[kb + half * 2 + 1];
    bf.x = Wout[(size_t)(kb + half * 2 + 0) * CC + ncol];
    bf.y = Wout[(size_t)(kb + half * 2 + 1) * CC + ncol];
    acc = __builtin_amdgcn_wmma_f32_16x16x4_f32(
        false, a, false, bf, (short)0, acc, false, false);
  }

  float bias = bout[ncol];
  #pragma unroll
  for (int r = 0; r < 8; ++r) {
    int mrow = tm * 16 + r + half * 8;
    out[(size_t)mrow * CC + ncol] = acc[r] + bias;
  }
}

// ---------------------------------------------------------------------------
extern "C" void kernel_launch(void* const* d_in, const int* in_sizes, int n_in,
                              void* d_out, int out_size, void* d_ws, size_t ws_size,
                              hipStream_t stream) {
  const float* feat = (const float*)d_in[0];  // [128,256,64,64]
  const float* ref  = (const float*)d_in[1];  // [128,17,2]
  const float* Woff = (const float*)d_in[2];  // [256,64]
  const float* boff = (const float*)d_in[3];  // [64]
  const float* Ww   = (const float*)d_in[4];  // [256,32]
  const float* bw   = (const float*)d_in[5];  // [32]
  const float* Wout = (const float*)d_in[6];  // [256,256]
  const float* bout = (const float*)d_in[7];  // [256]
  float* out = (float*)d_out;                 // [128,17,256] = [2176,256]

  float* ws   = (float*)d_ws;
  float* q    = ws;                        // 2176*256
  float* S    = q    + (size_t)ROWS * CC;  // 2176*96
  float* grid = S    + (size_t)ROWS * 96;  // 2176*64
  float* wts  = grid + (size_t)ROWS * 64;  // 2176*32
  float* agg  = wts  + (size_t)ROWS * HP;  // 2176*256

  k_sample_q   <<<ROWS, CC, 0, stream>>>(feat, ref, q);
  k_qproj_wmma <<<dim3(ROWS / 16, 6), 32, 0, stream>>>(q, Woff, boff, Ww, bw, S);
  k_softmax_grid<<<ROWS / 8, 256, 0, stream>>>(S, ref, grid, wts);
  k_deform_agg <<<ROWS, CC, 0, stream>>>(feat, grid, wts, agg);
  k_outproj_wmma<<<dim3(ROWS / 16, CC / 16), 32, 0, stream>>>(agg, Wout, bout, out);
}